// CabbageHeadRefinementLoss_60876866453863
// MI455X (gfx1250) — compile-verified
//
#include <hip/hip_runtime.h>
#include <hip/hip_bf16.h>

// ---------------------------------------------------------------------------
// CabbageHeadRefinementLoss for MI455X (gfx1250)
// Dominant cost: N x N ball-query (rank-3 Gram matrix) -> V_WMMA_F32_16X16X4_F32
// d2_ij = nrm_i + nrm_j - 2*dot(p_i,p_j) computed as A(16x4) x B(4x16) + C:
//   A row i  = [x, y, z, nrm_i]
//   B col j  = [-2x, -2y, -2z, 1]
//   C[:, j]  = nrm_j
// ---------------------------------------------------------------------------

#define BATCH  2
#define N_PTS  8192
#define ACCSZ  32

#define W_REF 0.3f
#define W_CON 0.2f
#define W_BND 2.0f
#define W_SHP 0.5f
#define W_SMO 0.3f
#define W_SIZ 0.8f
#define W_CNN 0.6f
#define R2C   0.0025f   // 0.05^2

typedef float v2f __attribute__((ext_vector_type(2)));
typedef float v8f __attribute__((ext_vector_type(8)));

// ---------------------------------------------------------------------------
// K0: zero per-batch accumulators
// acc layout (per batch, 32 floats):
//  [0] sum nll*w   [1] sum consist  [2] n_pred  [3] n_gt
//  [4..6] sum pred*xyz  [7] n_boundary
//  [8..13] cov sums xx,xy,xz,yy,yz,zz  [14] sum d*m  [15] sum d^2*m
//  [16] max(d*m) (float bits via uint atomicMax)  [17] smooth_num  [18] smooth_den
// ---------------------------------------------------------------------------
__global__ void k0_init(float* __restrict__ acc) {
  int t = threadIdx.x;
  if (t < BATCH * ACCSZ) acc[t] = 0.0f;
}

__device__ __forceinline__ float waveSum(float v) {
  #pragma unroll
  for (int off = 16; off > 0; off >>= 1) v += __shfl_xor(v, off, 32);
  return v;
}

// ---------------------------------------------------------------------------
// K1: per-point pass: softmax / CE / boundary / argmax, pack column data
// ---------------------------------------------------------------------------
__global__ void __launch_bounds__(256)
k1_pointpass(const float* __restrict__ logits, const float* __restrict__ olog,
             const float* __restrict__ hprob, const int* __restrict__ tgt,
             const float* __restrict__ pts, float* __restrict__ colpack,
             float* __restrict__ ph, float* __restrict__ acc) {
  const int nb = N_PTS / 256;
  int b = blockIdx.x / nb;
  int i = (blockIdx.x % nb) * 256 + threadIdx.x;
  size_t gi = (size_t)b * N_PTS + i;

  const float* l = logits + gi * 3;
  float l0 = l[0], l1 = l[1], l2 = l[2];
  float mx = fmaxf(l0, fmaxf(l1, l2));
  float e0 = __expf(l0 - mx), e1 = __expf(l1 - mx), e2 = __expf(l2 - mx);
  float s = e0 + e1 + e2, inv = 1.0f / s, ls = __logf(s);
  float p0 = e0 * inv, p1 = e1 * inv, p2 = e2 * inv;

  int t = tgt[gi];
  float lt = (t == 0) ? l0 : ((t == 1) ? l1 : l2);
  float nll = -(lt - mx - ls);

  float h = hprob[gi];
  float bm = (h > 0.3f && h < 0.7f) ? 1.0f : 0.0f;
  float wgt = 1.0f + (W_BND - 1.0f) * bm;

  const float* o = olog + gi * 3;
  float o0 = o[0], o1 = o[1], o2 = o[2];
  float omx = fmaxf(o0, fmaxf(o1, o2));
  float f0 = __expf(o0 - omx), f1 = __expf(o1 - omx), f2 = __expf(o2 - omx);
  float oinv = 1.0f / (f0 + f1 + f2);
  float q0 = f0 * oinv, q1 = f1 * oinv, q2 = f2 * oinv;
  float cons = (p0 - q0) * (p0 - q0) + (p1 - q1) * (p1 - q1) + (p2 - q2) * (p2 - q2);

  float phv = (l2 > l0 && l2 > l1) ? 1.0f : 0.0f;  // argmax==2 (first-max tie rule)
  float gt  = (t == 2) ? 1.0f : 0.0f;

  const float* P = pts + gi * 3;
  float x = P[0], y = P[1], z = P[2];
  float nrm = x * x + y * y + z * z;

  float* cp = colpack + gi * 6;
  cp[0] = -2.0f * x; cp[1] = -2.0f * y; cp[2] = -2.0f * z;
  cp[3] = nrm;       cp[4] = bm;        cp[5] = p2;
  ph[gi] = phv;

  float* ab = acc + b * ACCSZ;
  float vals[8] = { nll * wgt, cons, phv, gt, phv * x, phv * y, phv * z, bm };
  #pragma unroll
  for (int k = 0; k < 8; k++) {
    float v = waveSum(vals[k]);
    if ((threadIdx.x & 31) == 0) atomicAdd(&ab[k], v);
  }
}

// ---------------------------------------------------------------------------
// K2: masked covariance + connectivity stats (needs center from K1)
// ---------------------------------------------------------------------------
__global__ void __launch_bounds__(256)
k2_geo(const float* __restrict__ pts, const float* __restrict__ ph,
       float* __restrict__ acc) {
  const int nb = N_PTS / 256;
  int b = blockIdx.x / nb;
  int i = (blockIdx.x % nb) * 256 + threadIdx.x;
  size_t gi = (size_t)b * N_PTS + i;

  float* ab = acc + b * ACCSZ;
  float n  = ab[2];
  float nz = fmaxf(n, 1.0f);
  float cx = ab[4] / nz, cy = ab[5] / nz, cz = ab[6] / nz;

  const float* P = pts + gi * 3;
  float m = ph[gi];
  float dx = P[0] - cx, dy = P[1] - cy, dz = P[2] - cz;
  float ax = dx * m, ay = dy * m, az = dz * m;
  float d = sqrtf(dx * dx + dy * dy + dz * dz + 1e-12f);

  float vals[8] = { ax * ax, ax * ay, ax * az, ay * ay, ay * az, az * az,
                    d * m, d * d * m };
  #pragma unroll
  for (int k = 0; k < 8; k++) {
    float v = waveSum(vals[k]);
    if ((threadIdx.x & 31) == 0) atomicAdd(&ab[8 + k], v);
  }
  float dm = d * m;
  #pragma unroll
  for (int off = 16; off > 0; off >>= 1) dm = fmaxf(dm, __shfl_xor(dm, off, 32));
  if ((threadIdx.x & 31) == 0)
    atomicMax((unsigned int*)&ab[16], __float_as_uint(dm));  // dm >= 0
}

// ---------------------------------------------------------------------------
// K3: surface-smoothness ball query via WMMA f32 16x16x4.
// 8 waves/WG, wave owns one 16-row tile; columns staged in LDS in 24KB chunks.
// ---------------------------------------------------------------------------
#define CHUNK 1024
__global__ void __launch_bounds__(256)
k3_smooth(const float* __restrict__ colpack, float* __restrict__ acc) {
  __shared__ float sc[CHUNK * 6];
  const int wgPerBatch = (N_PTS / 16) / 8;      // 64
  int b    = blockIdx.x / wgPerBatch;
  int wg   = blockIdx.x % wgPerBatch;
  int wave = threadIdx.x >> 5;
  int lane = threadIdx.x & 31;
  int rt   = wg * 8 + wave;                     // row tile 0..511
  int lrow = lane & 15;
  int i    = rt * 16 + lrow;

  const float* colB = colpack + (size_t)b * N_PTS * 6;
  const float* cpA  = colB + (size_t)i * 6;

  // A (16x4 f32): lanes 0-15 -> (x, y); lanes 16-31 -> (z, nrm). (-0.5 undoes -2 pack)
  v2f a;
  if (lane < 16) { a.x = -0.5f * cpA[0]; a.y = -0.5f * cpA[1]; }
  else           { a.x = -0.5f * cpA[2]; a.y = cpA[3]; }

  float cnt[8], s1[8], s2[8];
  #pragma unroll
  for (int v = 0; v < 8; v++) { cnt[v] = 0.0f; s1[v] = 0.0f; s2[v] = 0.0f; }

  for (int cc = 0; cc < N_PTS / CHUNK; cc++) {
    __syncthreads();
    for (int k = threadIdx.x; k < CHUNK * 6; k += 256)
      sc[k] = colB[(size_t)cc * CHUNK * 6 + k];
    __syncthreads();

    for (int ct = 0; ct < CHUNK / 16; ct++) {
      int base = (ct * 16 + lrow) * 6;
      float bx  = sc[base + 0], by = sc[base + 1], bz = sc[base + 2];
      float nr  = sc[base + 3], bmj = sc[base + 4], pj = sc[base + 5];
      // B (4x16 f32): lanes 0-15 -> (K0,K1) = (-2x,-2y); lanes 16-31 -> (K2,K3) = (-2z, 1)
      v2f bb;
      if (lane < 16) { bb.x = bx; bb.y = by; }
      else           { bb.x = bz; bb.y = 1.0f; }
      v8f c;
      #pragma unroll
      for (int v = 0; v < 8; v++) c[v] = nr;   // + nrm_j bias per column
      v8f d = __builtin_amdgcn_wmma_f32_16x16x4_f32(
          false, a, false, bb, (short)0, c, false, false);
      #pragma unroll
      for (int v = 0; v < 8; v++) {
        float w = (d[v] < R2C) ? bmj : 0.0f;
        cnt[v] += w;
        s1[v]  += w * pj;
        s2[v]  += w * pj * pj;
      }
    }
  }

  // reduce across the 16 lanes that share the same rows
  #pragma unroll
  for (int v = 0; v < 8; v++) {
    #pragma unroll
    for (int off = 1; off < 16; off <<= 1) {
      cnt[v] += __shfl_xor(cnt[v], off, 32);
      s1[v]  += __shfl_xor(s1[v],  off, 32);
      s2[v]  += __shfl_xor(s2[v],  off, 32);
    }
  }
  if ((lane & 15) == 0) {
    float num = 0.0f, den = 0.0f;
    int rbase = rt * 16 + ((lane >= 16) ? 8 : 0);
    #pragma unroll
    for (int v = 0; v < 8; v++) {
      float bmi  = colB[(size_t)(rbase + v) * 6 + 4];
      float C    = cnt[v];
      float varr = (s2[v] - s1[v] * s1[v] / fmaxf(C, 1.0f)) / fmaxf(C - 1.0f, 1.0f);
      float valid = (C > 1.0f) ? bmi : 0.0f;
      num += varr * valid;
      den += valid;
    }
    float* ab = acc + b * ACCSZ;
    atomicAdd(&ab[17], num);
    atomicAdd(&ab[18], den);
  }
}

// ---------------------------------------------------------------------------
// K4: analytic 3x3 symmetric eigenvalues + combine all terms
// ---------------------------------------------------------------------------
__device__ void eig3_sym(float xx, float xy, float xz, float yy, float yz,
                         float zz, float ev[3]) {
  float p1 = xy * xy + xz * xz + yz * yz;
  float q  = (xx + yy + zz) / 3.0f;
  float p2 = (xx - q) * (xx - q) + (yy - q) * (yy - q) + (zz - q) * (zz - q) + 2.0f * p1;
  float p  = sqrtf(fmaxf(p2, 0.0f) / 6.0f);
  if (p < 1e-20f) { ev[0] = ev[1] = ev[2] = q; return; }
  float inv = 1.0f / p;
  float bxx = (xx - q) * inv, byy = (yy - q) * inv, bzz = (zz - q) * inv;
  float bxy = xy * inv, bxz = xz * inv, byz = yz * inv;
  float detB = bxx * (byy * bzz - byz * byz) - bxy * (bxy * bzz - byz * bxz)
             + bxz * (bxy * byz - byy * bxz);
  float r = fminf(1.0f, fmaxf(-1.0f, 0.5f * detB));
  float phi = acosf(r) / 3.0f;
  float e2 = q + 2.0f * p * cosf(phi);
  float e0 = q + 2.0f * p * cosf(phi + 2.0943951023931953f);
  ev[0] = e0; ev[1] = 3.0f * q - e2 - e0; ev[2] = e2;
}

__global__ void k4_final(const float* __restrict__ acc, float* __restrict__ out) {
  __shared__ float tot[BATCH];
  int b = threadIdx.x;
  if (b < BATCH) {
    const float* ab = acc + b * ACCSZ;
    float refinement  = ab[0] / (float)N_PTS;
    float consistency = ab[1] / (float)(N_PTS * 3);
    float n = ab[2], gtn = ab[3];
    float nz = fmaxf(n, 1.0f);

    float ev[3];
    eig3_sym(ab[8] / nz, ab[9] / nz, ab[10] / nz,
             ab[11] / nz, ab[12] / nz, ab[13] / nz, ev);
    float ainv = 1.0f / (ev[2] + 1e-8f);
    float t1 = ev[1] * ainv - 1.0f, t0 = ev[0] * ainv - 1.0f;
    float shape = (n >= 10.0f) ? (t1 * t1 + t0 * t0) : 0.0f;

    float sdm = ab[14], sd2m = ab[15];
    float varr = (sd2m - sdm * sdm / nz) / fmaxf(n - 1.0f, 1.0f);
    float maxd = ab[16];  // stored as float bits via uint atomicMax (>=0)
    float conn = (n >= 5.0f) ? varr / (maxd + 1e-8f) : 0.0f;

    float smooth = (ab[7] >= 5.0f) ? ab[17] / fmaxf(ab[18], 1.0f) : 0.0f;

    float dv = n - gtn;
    float vol = dv * dv;
    float rel = fabsf(dv) / fmaxf(gtn, 1.0f);
    float size = (gtn > 0.0f) ? (vol + 0.5f * rel) : vol;

    tot[b] = W_REF * refinement + W_CON * consistency + W_SHP * shape
           + W_SMO * smooth + W_SIZ * size + W_CNN * conn;
  }
  __syncthreads();
  if (threadIdx.x == 0) {
    float s = 0.0f;
    for (int k = 0; k < BATCH; k++) s += tot[k];
    out[0] = s / (float)BATCH;
  }
}

// ---------------------------------------------------------------------------
extern "C" void kernel_launch(void* const* d_in, const int* in_sizes, int n_in,
                              void* d_out, int out_size, void* d_ws, size_t ws_size,
                              hipStream_t stream) {
  const float* logits = (const float*)d_in[0];
  const float* olog   = (const float*)d_in[1];
  const float* hprob  = (const float*)d_in[2];
  const int*   tgt    = (const int*)d_in[3];
  const float* pts    = (const float*)d_in[4];

  float* ws      = (float*)d_ws;                       // needs ~460KB
  float* colpack = ws;                                 // [B][N][6]
  float* ph      = ws + (size_t)BATCH * N_PTS * 6;     // [B][N]
  float* acc     = ws + (size_t)BATCH * N_PTS * 7;     // [B][ACCSZ]
  float* out     = (float*)d_out;

  k0_init<<<1, 64, 0, stream>>>(acc);
  k1_pointpass<<<BATCH * (N_PTS / 256), 256, 0, stream>>>(
      logits, olog, hprob, tgt, pts, colpack, ph, acc);
  k2_geo<<<BATCH * (N_PTS / 256), 256, 0, stream>>>(pts, ph, acc);
  k3_smooth<<<BATCH * ((N_PTS / 16) / 8), 256, 0, stream>>>(colpack, acc);
  k4_final<<<1, 32, 0, stream>>>(acc, out);
}